// MaskBfEstimator_88983132438899
// MI455X (gfx1250) — compile-verified
//
#include <hip/hip_runtime.h>
#include <hip/hip_bf16.h>

#define FRAMES   4000
#define MICS     8
#define BINS     513
#define POWM_ITR 5
#define FSTRIDE  (MICS * BINS)   // 4104 floats per frame of x
#define MSTRIDE  (2 * BINS)      // 1026 floats per frame of masks

typedef __attribute__((ext_vector_type(2))) float v2f;
typedef __attribute__((ext_vector_type(8))) float v8f;
typedef __attribute__((ext_vector_type(4))) unsigned int u32x4;
typedef __attribute__((ext_vector_type(8))) int i32x8;
typedef __attribute__((ext_vector_type(4))) int i32x4;

// ---------------------------------------------------------------------------
// TDM helper: 3D tensor tile load Global -> LDS  (D# per CDNA5 ISA ch.8)
//   dim0 = contiguous elements (bins), dim1 stride = s0, dim2 stride = s1.
//   data_size = 4 bytes, tile z-extent = 16 frames, no padding, no multicast.
//   6-arg builtin form: (g0, g1, g2, g3, <unused group>, cpol).
// ---------------------------------------------------------------------------
__device__ __forceinline__ void tdm_load_3d(unsigned lds_byte_off,
                                            const float* gptr,
                                            unsigned tdim0, unsigned tdim1,
                                            unsigned tile0, unsigned tile1,
                                            unsigned tile2,
                                            unsigned long long s0,
                                            unsigned long long s1) {
  unsigned long long ga = (unsigned long long)(uintptr_t)gptr;
  u32x4 g0;
  g0[0] = 1u;                                          // count=1, user D#
  g0[1] = lds_byte_off;                                // lds_addr
  g0[2] = (unsigned)(ga & 0xFFFFFFFFu);                // global_addr[31:0]
  g0[3] = (unsigned)((ga >> 32) & 0x1FFFFFFu) | (2u << 30);  // addr[56:32], type=2
  i32x8 g1;
  g1[0] = (int)(2u << 16);                             // data_size=4B, wg_mask=0
  g1[1] = (int)((tdim0 & 0xFFFFu) << 16);              // tensor_dim0[15:0]
  g1[2] = (int)((tdim0 >> 16) | ((tdim1 & 0xFFFFu) << 16));
  g1[3] = (int)((tdim1 >> 16) | (tile0 << 16));        // tile_dim0
  g1[4] = (int)(tile1 | (tile2 << 16));                // tile_dim1, tile_dim2
  g1[5] = (int)(s0 & 0xFFFFFFFFu);                     // tensor_dim0_stride lo
  g1[6] = (int)(((s0 >> 32) & 0xFFFFu) | ((s1 & 0xFFFFu) << 16));
  g1[7] = (int)((s1 >> 16) & 0xFFFFFFFFu);             // tensor_dim1_stride hi
  i32x4 g2;
  g2[0] = 16;                                          // tensor_dim2 (z extent)
  g2[1] = 0; g2[2] = 0; g2[3] = 0;                     // no dim3 / iterate
  i32x4 g3 = {0, 0, 0, 0};                             // dims 3-4 unused
  i32x8 g4 = {0, 0, 0, 0, 0, 0, 0, 0};                 // unused trailing group
  __builtin_amdgcn_tensor_load_to_lds(g0, g1, g2, g3, g4, 0);
}

// ---------------------------------------------------------------------------
// Kernel A: mask variance -> per-(mask,bin) scale = 1/(FRAMES*sqrt(var))
// ---------------------------------------------------------------------------
__global__ void k_maskvar(const float* __restrict__ masks,
                          float* __restrict__ varscale) {
  int j = blockIdx.x * blockDim.x + threadIdx.x;   // j in [0, 1026)
  if (j >= MSTRIDE) return;
  float acc = 0.f;
  for (int f = 0; f < FRAMES; ++f) {
    float m = masks[(size_t)f * MSTRIDE + j];
    acc += m * m;                                  // |m|^2 == m^2
  }
  float var = acc / (float)FRAMES;
  varscale[j] = 1.0f / ((float)FRAMES * sqrtf(var));
}

// ---------------------------------------------------------------------------
// Kernel B: per-bin 16x16 f32 covariance via V_WMMA_F32_16X16X4_F32, with
// TDM double-buffered LDS staging (TENSOR_LOAD_TO_LDS + s_wait_tensorcnt).
//   A rows 0..7 = Re(x_mic), rows 8..15 = Im(x_mic); K = frames.
//   acc = A * (|mask| .* A)^T accumulated over 4000 frames, two masks.
// Block: 256 threads (8 waves), 16 bins per block, 2 bins per wave.
// TDM LDS packing is [f][mic][b] (b fastest, 16 bins): addr = f*128+mic*16+b.
// ---------------------------------------------------------------------------
#define BTILE   16
#define FCH     16
#define NCH     (FRAMES / FCH)   // 250
#define XSR_OFF 0                // x_re tile: 2048 floats
#define XSI_OFF 2052             // x_im tile (+4 dw gap -> disjoint bank group)
#define MS_OFF  4104             // masks tile: 512 floats [f][k][b]
#define BUFSZ   4616             // floats per buffer (8B-aligned)

__global__ void __launch_bounds__(256)
k_cov(const float* __restrict__ x_re, const float* __restrict__ x_im,
      const float* __restrict__ masks, float* __restrict__ covraw) {
  __shared__ float lds[2 * BUFSZ];

  const int t    = threadIdx.x;
  const int b0   = blockIdx.x * BTILE;
  const int lane = t & 31;
  const int wv   = t >> 5;
  const int bloc = wv * 2;                 // this wave's two local bins

  const unsigned lds_base = (unsigned)(unsigned long long)(uintptr_t)(&lds[0]);
  const unsigned tdim0 = (unsigned)(BINS - b0);   // OOB bins -> zero fill

  v8f acc[2][2] = {};                      // [local bin][mask]

  const int Mi   = lane & 15;              // A row M == B col N
  const int mic  = Mi & 7;
  const int reim = Mi >> 3;                // 0: x_re half, 1: x_im half
  const int fof  = (lane >> 4) << 1;       // K offset per ISA A/B layout

  if (wv == 0) {                           // prologue: chunk 0 -> buffer 0
    tdm_load_3d(lds_base + XSR_OFF * 4u, x_re + b0, tdim0, 8, 16, 8, 16, 513ull, 4104ull);
    tdm_load_3d(lds_base + XSI_OFF * 4u, x_im + b0, tdim0, 8, 16, 8, 16, 513ull, 4104ull);
    tdm_load_3d(lds_base + MS_OFF * 4u, masks + b0, tdim0, 2, 16, 2, 16, 513ull, 1026ull);
  }

  for (int c = 0; c < NCH; ++c) {
    if (wv == 0) __builtin_amdgcn_s_wait_tensorcnt(0);   // chunk c landed
    __syncthreads();                                     // visible to all waves
    if (wv == 0 && c + 1 < NCH) {                        // prefetch chunk c+1
      const unsigned bb = ((unsigned)(c + 1) & 1u) * (BUFSZ * 4u);
      const size_t fo = (size_t)(c + 1) * FCH;
      tdm_load_3d(lds_base + bb + XSR_OFF * 4u, x_re + fo * FSTRIDE + b0,
                  tdim0, 8, 16, 8, 16, 513ull, 4104ull);
      tdm_load_3d(lds_base + bb + XSI_OFF * 4u, x_im + fo * FSTRIDE + b0,
                  tdim0, 8, 16, 8, 16, 513ull, 4104ull);
      tdm_load_3d(lds_base + bb + MS_OFF * 4u, masks + fo * MSTRIDE + b0,
                  tdim0, 2, 16, 2, 16, 513ull, 1026ull);
    }
    const float* buf  = &lds[(c & 1) * BUFSZ];
    const float* xbuf = buf + (reim ? XSI_OFF : XSR_OFF);
    const float* mbuf = buf + MS_OFF;
#pragma unroll
    for (int l2 = 0; l2 < 2; ++l2) {
      const int bl = bloc + l2;
#pragma unroll
      for (int k0 = 0; k0 < FCH; k0 += 4) {
        const int fl = k0 + fof;
        const int xa = fl * 128 + mic * 16 + bl;    // [f][mic][b]
        float a0 = xbuf[xa];
        float a1 = xbuf[xa + 128];                  // f+1 (2addr_b32 pair)
        const int ma = fl * 32 + bl;                // [f][k][b]
        float p0 = fabsf(mbuf[ma]),      p1 = fabsf(mbuf[ma + 32]);  // mask0
        float q0 = fabsf(mbuf[ma + 16]), q1 = fabsf(mbuf[ma + 48]);  // mask1
        v2f a   = {a0, a1};
        v2f bv0 = {a0 * p0, a1 * p1};               // B shares A lane mapping
        v2f bv1 = {a0 * q0, a1 * q1};
        acc[l2][0] = __builtin_amdgcn_wmma_f32_16x16x4_f32(
            false, a, false, bv0, (short)0, acc[l2][0], false, false);
        acc[l2][1] = __builtin_amdgcn_wmma_f32_16x16x4_f32(
            false, a, false, bv1, (short)0, acc[l2][1], false, false);
      }
    }
  }

  // Store raw 16x16 accumulators: lane l holds D[g + 8*(l>>4)][l&15], g=0..7
  const int col   = lane & 15;
  const int rbase = (lane >> 4) << 3;
#pragma unroll
  for (int l2 = 0; l2 < 2; ++l2) {
    int bin = b0 + bloc + l2;
    if (bin >= BINS) continue;
#pragma unroll
    for (int msk = 0; msk < 2; ++msk) {
      float* dst = covraw + (size_t)(bin * 2 + msk) * 256;
#pragma unroll
      for (int g = 0; g < 8; ++g) dst[(g + rbase) * 16 + col] = acc[l2][msk][g];
    }
  }
}

// ---------------------------------------------------------------------------
// Kernel C: per-bin power iteration. wcov built from 16x16 quadrants, no-pivot
// complex LU of wcov1 (HPD), 5 iterations of w = wcov1^{-1} wcov2 w.
// ---------------------------------------------------------------------------
__global__ void k_powm(const float* __restrict__ covraw,
                       const float* __restrict__ varscale,
                       float* __restrict__ wbuf) {
  int b = blockIdx.x * blockDim.x + threadIdx.x;
  if (b >= BINS) return;
  const float s0 = varscale[b];
  const float s1 = varscale[BINS + b];
  const float* c0 = covraw + (size_t)(b * 2 + 0) * 256;
  const float* c1 = covraw + (size_t)(b * 2 + 1) * 256;

  float Ar[64], Ai[64], Br[64], Bi[64];
  for (int i = 0; i < 8; ++i)
    for (int j = 0; j < 8; ++j) {
      int ij = i * 16 + j, i8j = (i + 8) * 16 + j;
      int ij8 = i * 16 + j + 8, i8j8 = (i + 8) * 16 + j + 8;
      Ar[i * 8 + j] = (c0[ij] + c0[i8j8]) * s0;   // Re(wcov1)
      Ai[i * 8 + j] = (c0[i8j] - c0[ij8]) * s0;   // Im(wcov1)
      Br[i * 8 + j] = (c1[ij] + c1[i8j8]) * s1;   // Re(wcov2)
      Bi[i * 8 + j] = (c1[i8j] - c1[ij8]) * s1;   // Im(wcov2)
    }

  for (int k = 0; k < 8; ++k) {                   // in-place complex LU
    float dr = Ar[k * 8 + k], di = Ai[k * 8 + k];
    float inv = 1.0f / (dr * dr + di * di);
    float pr = dr * inv, pi = -di * inv;
    for (int i = k + 1; i < 8; ++i) {
      float xr = Ar[i * 8 + k], xi = Ai[i * 8 + k];
      float lr = xr * pr - xi * pi;
      float li = xr * pi + xi * pr;
      Ar[i * 8 + k] = lr; Ai[i * 8 + k] = li;
      for (int j = k + 1; j < 8; ++j) {
        float ur = Ar[k * 8 + j], ui = Ai[k * 8 + j];
        Ar[i * 8 + j] -= lr * ur - li * ui;
        Ai[i * 8 + j] -= lr * ui + li * ur;
      }
    }
  }

  float wr[8] = {0}, wi[8] = {0};
  wr[0] = 1.0f;
  for (int it = 0; it < POWM_ITR; ++it) {
    float nrm = 0.f;
    for (int i = 0; i < 8; ++i) nrm += wr[i] * wr[i] + wi[i] * wi[i];
    float invn = 1.0f / sqrtf(nrm);
    for (int i = 0; i < 8; ++i) { wr[i] *= invn; wi[i] *= invn; }
    float vr[8], vi[8];
    for (int i = 0; i < 8; ++i) {                 // v = wcov2 * w
      float sr = 0.f, si = 0.f;
      for (int j = 0; j < 8; ++j) {
        sr += Br[i * 8 + j] * wr[j] - Bi[i * 8 + j] * wi[j];
        si += Br[i * 8 + j] * wi[j] + Bi[i * 8 + j] * wr[j];
      }
      vr[i] = sr; vi[i] = si;
    }
    for (int i = 0; i < 8; ++i)                   // forward solve L y = v
      for (int j = 0; j < i; ++j) {
        vr[i] -= Ar[i * 8 + j] * vr[j] - Ai[i * 8 + j] * vi[j];
        vi[i] -= Ar[i * 8 + j] * vi[j] + Ai[i * 8 + j] * vr[j];
      }
    for (int i = 7; i >= 0; --i) {                // back solve U w = y
      float sr = vr[i], si = vi[i];
      for (int j = i + 1; j < 8; ++j) {
        sr -= Ar[i * 8 + j] * wr[j] - Ai[i * 8 + j] * wi[j];
        si -= Ar[i * 8 + j] * wi[j] + Ai[i * 8 + j] * wr[j];
      }
      float dr = Ar[i * 8 + i], di = Ai[i * 8 + i];
      float inv = 1.0f / (dr * dr + di * di);
      wr[i] = (sr * dr + si * di) * inv;
      wi[i] = (si * dr - sr * di) * inv;
    }
  }
  for (int i = 0; i < 8; ++i) {
    wbuf[b * 16 + 2 * i]     = wr[i];
    wbuf[b * 16 + 2 * i + 1] = wi[i];
  }
}

// ---------------------------------------------------------------------------
// Kernel D: y[b,f] = w^H x; partial per-(frame-chunk,bin) cov/pwr reductions.
// ---------------------------------------------------------------------------
__global__ void k_apply(const float* __restrict__ x_re,
                        const float* __restrict__ x_im,
                        const float* __restrict__ s_re,
                        const float* __restrict__ s_im,
                        const float* __restrict__ wbuf,
                        float* __restrict__ ybuf,
                        float* __restrict__ redp) {
  int b = blockIdx.x * 64 + threadIdx.x;
  bool act = (b < BINS);
  int bb = act ? b : (BINS - 1);
  float wr[8], wi[8];
#pragma unroll
  for (int i = 0; i < 8; ++i) {
    wr[i] = wbuf[bb * 16 + 2 * i];
    wi[i] = wbuf[bb * 16 + 2 * i + 1];
  }
  const int f0 = blockIdx.y * (FRAMES / 50);      // 80 frames per chunk
  float cr = 0.f, ci = 0.f, pw = 0.f;
  for (int f = f0; f < f0 + FRAMES / 50; ++f) {
    const float* xr = x_re + (size_t)f * FSTRIDE + bb;
    const float* xi = x_im + (size_t)f * FSTRIDE + bb;
    float yr = 0.f, yi = 0.f;
#pragma unroll
    for (int m = 0; m < 8; ++m) {                 // conj(w) . x
      float ar = xr[m * BINS], ai = xi[m * BINS];
      yr += wr[m] * ar + wi[m] * ai;
      yi += wr[m] * ai - wi[m] * ar;
    }
    if (act) {
      size_t yo = 2 * ((size_t)f * BINS + b);
      ybuf[yo] = yr; ybuf[yo + 1] = yi;
    }
    float sr = s_re[(size_t)f * FSTRIDE + bb];    // target channel 0
    float si = s_im[(size_t)f * FSTRIDE + bb];
    cr += sr * yr + si * yi;                      // s * conj(y)
    ci += si * yr - sr * yi;
    pw += yr * yr + yi * yi;
  }
  if (act) {
    size_t ro = ((size_t)blockIdx.y * BINS + b) * 3;
    redp[ro] = cr; redp[ro + 1] = ci; redp[ro + 2] = pw;
  }
}

// Kernel D2: reduce the 50 partials per bin (deterministic, no atomics).
__global__ void k_red(const float* __restrict__ redp, float* __restrict__ red) {
  int b = blockIdx.x * blockDim.x + threadIdx.x;
  if (b >= BINS) return;
  float cr = 0.f, ci = 0.f, pw = 0.f;
  for (int c = 0; c < 50; ++c) {
    size_t ro = ((size_t)c * BINS + b) * 3;
    cr += redp[ro]; ci += redp[ro + 1]; pw += redp[ro + 2];
  }
  red[b * 4]     = cr;
  red[b * 4 + 1] = ci;
  red[b * 4 + 2] = pw;
}

// ---------------------------------------------------------------------------
// Kernel E: out[f,0,b] = y[b,f] * (cov[b]/pwr[b]); complex64 as float pairs.
// ---------------------------------------------------------------------------
__global__ void k_scale(const float* __restrict__ ybuf,
                        const float* __restrict__ red,
                        float* __restrict__ out) {
  size_t g = (size_t)blockIdx.x * blockDim.x + threadIdx.x;
  if (g >= (size_t)FRAMES * BINS) return;
  int b = (int)(g % BINS);
  float inv = 1.0f / red[b * 4 + 2];
  float fr = red[b * 4] * inv, fi = red[b * 4 + 1] * inv;
  float yr = ybuf[2 * g], yi = ybuf[2 * g + 1];
  out[2 * g]     = yr * fr - yi * fi;
  out[2 * g + 1] = yr * fi + yi * fr;
}

// ---------------------------------------------------------------------------
extern "C" void kernel_launch(void* const* d_in, const int* in_sizes, int n_in,
                              void* d_out, int out_size, void* d_ws, size_t ws_size,
                              hipStream_t stream) {
  const float* x_re  = (const float*)d_in[0];
  const float* x_im  = (const float*)d_in[1];
  const float* s_re  = (const float*)d_in[2];
  const float* s_im  = (const float*)d_in[3];
  const float* masks = (const float*)d_in[4];
  float* out = (float*)d_out;

  float* ws       = (float*)d_ws;
  float* varscale = ws;                                   // 1026
  float* wbuf     = varscale + MSTRIDE;                   // 513*16 = 8208
  float* red      = wbuf + BINS * 16;                     // 513*4  = 2052
  float* redp     = red + BINS * 4;                       // 50*513*3 = 76950
  float* covraw   = redp + 50 * BINS * 3;                 // 513*2*256 = 262656
  float* ybuf     = covraw + (size_t)BINS * 2 * 256;      // 4000*513*2 floats

  k_maskvar<<<(MSTRIDE + 255) / 256, 256, 0, stream>>>(masks, varscale);
  k_cov<<<(BINS + BTILE - 1) / BTILE, 256, 0, stream>>>(x_re, x_im, masks, covraw);
  k_powm<<<(BINS + 63) / 64, 64, 0, stream>>>(covraw, varscale, wbuf);
  k_apply<<<dim3((BINS + 63) / 64, 50), 64, 0, stream>>>(x_re, x_im, s_re, s_im,
                                                         wbuf, ybuf, redp);
  k_red<<<(BINS + 255) / 256, 256, 0, stream>>>(redp, red);
  size_t tot = (size_t)FRAMES * BINS;
  k_scale<<<(unsigned)((tot + 255) / 256), 256, 0, stream>>>(ybuf, red, out);
}